// NetAmazon_GAT_41102837023183
// MI455X (gfx1250) — compile-verified
//
#include <hip/hip_runtime.h>
#include <math.h>

typedef __attribute__((ext_vector_type(2))) float v2f;
typedef __attribute__((ext_vector_type(8))) float v8f;

#define NEG_SLOPE 0.2f
#define GAT_EPS   1e-16f

// ---------------- helpers ----------------
__device__ __forceinline__ void atomicMaxF(float* addr, float val) {
  // sign-split trick: positive floats order as ints, negative as reversed uints
  if (val >= 0.0f) atomicMax((int*)addr, __float_as_int(val));
  else             atomicMin((unsigned int*)addr, __float_as_uint(val));
}

// ---------------- kernels ----------------

// Compact int edge arrays + append self-loops (edges [E, E+N) are loops).
__global__ void k_prep_edges(const int* __restrict__ ei, int* __restrict__ src,
                             int* __restrict__ dst, int E, int Et) {
  int i = blockIdx.x * blockDim.x + threadIdx.x;
  if (i >= Et) return;
  if (i < E) { src[i] = ei[i]; dst[i] = ei[E + i]; }
  else       { int n = i - E; src[i] = n; dst[i] = n; }
}

// FP32 WMMA GEMM: C[M,NCOL] = A[M,K] * B[K,NCOL].  One wave per 16x16 tile,
// 4 M-tiles per block sharing one N-tile (B fragments reused via cache).
// ISA 7.12.2 32-bit layouts: A 16x4 (lane<16: K=k,k+1 ; lane>=16: K=k+2,k+3),
// B 4x16 row-striped across lanes, C/D rows v + 8*(lane/16).
template<int K, int NCOL>
__global__ __launch_bounds__(128)
void k_gemm_wmma(const float* __restrict__ A, const float* __restrict__ B,
                 float* __restrict__ C, int Mtiles) {
  constexpr int NT = NCOL / 16;
  const int wave  = threadIdx.x >> 5;
  const int lane  = threadIdx.x & 31;
  const int tileN = blockIdx.x % NT;
  const int tileM = (blockIdx.x / NT) * 4 + wave;
  if (tileM >= Mtiles) return;               // wave-uniform: EXEC stays all-1s
  const int half = lane >> 4;
  const int lr   = lane & 15;
  const float* Ap = A + (size_t)(tileM * 16 + lr) * K + half * 2;
  const float* Bp = B + (size_t)(half * 2) * NCOL + tileN * 16 + lr;
  v8f acc = {};
  #pragma unroll 4
  for (int k = 0; k < K; k += 4) {
    v2f a = *(const v2f*)Ap;                 // 8B-aligned (K, k, half*2 all even)
    v2f b; b.x = Bp[0]; b.y = Bp[NCOL];
    acc = __builtin_amdgcn_wmma_f32_16x16x4_f32(false, a, false, b,
                                                (short)0, acc, false, false);
    Ap += 4;
    Bp += 4 * NCOL;
  }
  float* Cp = C + (size_t)(tileM * 16 + half * 8) * NCOL + tileN * 16 + lr;
  #pragma unroll
  for (int v = 0; v < 8; ++v) Cp[(size_t)v * NCOL] = acc[v];
}

// Per-(node,head) attention logits; also init segment max/sum.
template<int H, int C>
__global__ void k_attn_logits(const float* __restrict__ Hf,
                              const float* __restrict__ asrc,
                              const float* __restrict__ adst,
                              float* __restrict__ as, float* __restrict__ ad,
                              float* __restrict__ m, float* __restrict__ s, int N) {
  int idx = blockIdx.x * blockDim.x + threadIdx.x;
  if (idx >= N * H) return;
  int h = idx % H, n = idx / H;
  const float* hp = Hf + (size_t)n * (H * C) + h * C;
  float vs = 0.f, vd = 0.f;
  #pragma unroll
  for (int c = 0; c < C; ++c) {
    float x = hp[c];
    vs += x * asrc[h * C + c];
    vd += x * adst[h * C + c];
  }
  as[idx] = vs; ad[idx] = vd;
  m[idx] = -__builtin_inff();
  s[idx] = 0.f;
}

__global__ void k_zero(float* __restrict__ p, int n) {
  int i = blockIdx.x * blockDim.x + threadIdx.x;
  if (i < n) p[i] = 0.f;
}

template<int H>
__global__ void k_edge_max(const int* __restrict__ src, const int* __restrict__ dst,
                           const float* __restrict__ as, const float* __restrict__ ad,
                           float* __restrict__ m, int Et) {
  int idx = blockIdx.x * blockDim.x + threadIdx.x;
  if (idx >= Et * H) return;
  int h = idx % H, i = idx / H;
  int sN = src[i], dN = dst[i];
  float e = as[sN * H + h] + ad[dN * H + h];
  e = (e > 0.f) ? e : NEG_SLOPE * e;
  atomicMaxF(&m[dN * H + h], e);
}

template<int H>
__global__ void k_edge_sum(const int* __restrict__ src, const int* __restrict__ dst,
                           const float* __restrict__ as, const float* __restrict__ ad,
                           const float* __restrict__ m, float* __restrict__ s, int Et) {
  int idx = blockIdx.x * blockDim.x + threadIdx.x;
  if (idx >= Et * H) return;
  int h = idx % H, i = idx / H;
  int sN = src[i], dN = dst[i];
  float e = as[sN * H + h] + ad[dN * H + h];
  e = (e > 0.f) ? e : NEG_SLOPE * e;
  float w = __expf(e - m[dN * H + h]);
  atomicAdd(&s[dN * H + h], w);
}

template<int H, int C>
__global__ void k_edge_aggr(const int* __restrict__ src, const int* __restrict__ dst,
                            const float* __restrict__ as, const float* __restrict__ ad,
                            const float* __restrict__ m, const float* __restrict__ s,
                            const float* __restrict__ Hf, float* __restrict__ out, int Et) {
  int idx = blockIdx.x * blockDim.x + threadIdx.x;
  if (idx >= Et * H) return;
  int h = idx % H, i = idx / H;
  int sN = src[i], dN = dst[i];
  float e = as[sN * H + h] + ad[dN * H + h];
  e = (e > 0.f) ? e : NEG_SLOPE * e;
  float alpha = __expf(e - m[dN * H + h]) / (s[dN * H + h] + GAT_EPS);
  const float* hp = Hf + (size_t)sN * (H * C) + h * C;
  float*       op = out + (size_t)dN * (H * C) + h * C;
  #pragma unroll
  for (int c = 0; c < C; ++c) atomicAdd(&op[c], alpha * hp[c]);
}

template<int HC>
__global__ void k_bias_relu(float* __restrict__ out, const float* __restrict__ b, int N) {
  int idx = blockIdx.x * blockDim.x + threadIdx.x;
  if (idx >= N * HC) return;
  float v = out[idx] + b[idx % HC];
  out[idx] = fmaxf(v, 0.f);
}

// Layer-3 projection: [N,64] x [64,10] — too skinny for WMMA tiles; scalar.
__global__ void k_gemm_small(const float* __restrict__ A, const float* __restrict__ B,
                             float* __restrict__ C, int N, int K, int NC) {
  int idx = blockIdx.x * blockDim.x + threadIdx.x;
  if (idx >= N * NC) return;
  int c = idx % NC, n = idx / NC;
  const float* a = A + (size_t)n * K;
  float acc = 0.f;
  for (int k = 0; k < K; ++k) acc += a[k] * B[k * NC + c];
  C[idx] = acc;
}

__global__ void k_final_logsoftmax(const float* __restrict__ out3,
                                   const float* __restrict__ b,
                                   float* __restrict__ y, int N) {
  int n = blockIdx.x * blockDim.x + threadIdx.x;
  if (n >= N) return;
  float v[10];
  float mx = -__builtin_inff();
  #pragma unroll
  for (int c = 0; c < 10; ++c) { v[c] = out3[n * 10 + c] + b[c]; mx = fmaxf(mx, v[c]); }
  float sum = 0.f;
  #pragma unroll
  for (int c = 0; c < 10; ++c) sum += __expf(v[c] - mx);
  float l = logf(sum);
  #pragma unroll
  for (int c = 0; c < 10; ++c) y[n * 10 + c] = v[c] - mx - l;
}

// ---------------- launch ----------------
extern "C" void kernel_launch(void* const* d_in, const int* in_sizes, int n_in,
                              void* d_out, int out_size, void* d_ws, size_t ws_size,
                              hipStream_t stream) {
  const float* x      = (const float*)d_in[0];
  const int*   eidx   = (const int*)  d_in[1];
  const float* W1     = (const float*)d_in[2];
  const float* a1s    = (const float*)d_in[3];
  const float* a1d    = (const float*)d_in[4];
  const float* b1     = (const float*)d_in[5];
  const float* W2     = (const float*)d_in[6];
  const float* a2s    = (const float*)d_in[7];
  const float* a2d    = (const float*)d_in[8];
  const float* b2     = (const float*)d_in[9];
  const float* W3     = (const float*)d_in[10];
  const float* a3s    = (const float*)d_in[11];
  const float* a3d    = (const float*)d_in[12];
  const float* b3     = (const float*)d_in[13];
  float* y = (float*)d_out;

  const int F  = 300;
  const int E  = in_sizes[1] / 2;
  const int N  = in_sizes[0] / F;
  const int Et = E + N;                 // with self-loops
  const int Mt = N / 16;                // 50000/16 = 3125 exactly

  // workspace carve-out
  char* ws = (char*)d_ws;
  size_t off = 0;
  auto carve = [&](size_t bytes) -> void* {
    void* p = ws + off;
    off += (bytes + 255) & ~(size_t)255;
    return p;
  };
  int*   srcI = (int*)  carve((size_t)Et * 4);
  int*   dstI = (int*)  carve((size_t)Et * 4);
  float* bufH = (float*)carve((size_t)N * 256 * 4);
  float* bufO = (float*)carve((size_t)N * 256 * 4);
  float* aS   = (float*)carve((size_t)N * 16 * 4);
  float* aD   = (float*)carve((size_t)N * 16 * 4);
  float* mx   = (float*)carve((size_t)N * 16 * 4);
  float* sm   = (float*)carve((size_t)N * 16 * 4);

  const int TB = 256;
  auto nb = [](long long t, int b) { return (int)((t + b - 1) / b); };

  k_prep_edges<<<nb(Et, TB), TB, 0, stream>>>(eidx, srcI, dstI, E, Et);

  // ---- Layer 1: 300 -> 16 heads x 16 ch ----
  k_gemm_wmma<300, 256><<<((Mt + 3) / 4) * 16, 128, 0, stream>>>(x, W1, bufH, Mt);
  k_attn_logits<16, 16><<<nb((long long)N * 16, TB), TB, 0, stream>>>(bufH, a1s, a1d, aS, aD, mx, sm, N);
  k_zero<<<nb((long long)N * 256, TB), TB, 0, stream>>>(bufO, N * 256);
  k_edge_max<16><<<nb((long long)Et * 16, TB), TB, 0, stream>>>(srcI, dstI, aS, aD, mx, Et);
  k_edge_sum<16><<<nb((long long)Et * 16, TB), TB, 0, stream>>>(srcI, dstI, aS, aD, mx, sm, Et);
  k_edge_aggr<16, 16><<<nb((long long)Et * 16, TB), TB, 0, stream>>>(srcI, dstI, aS, aD, mx, sm, bufH, bufO, Et);
  k_bias_relu<256><<<nb((long long)N * 256, TB), TB, 0, stream>>>(bufO, b1, N);

  // ---- Layer 2: 256 -> 8 heads x 8 ch ----
  k_gemm_wmma<256, 64><<<((Mt + 3) / 4) * 4, 128, 0, stream>>>(bufO, W2, bufH, Mt);
  k_attn_logits<8, 8><<<nb((long long)N * 8, TB), TB, 0, stream>>>(bufH, a2s, a2d, aS, aD, mx, sm, N);
  k_zero<<<nb((long long)N * 64, TB), TB, 0, stream>>>(bufO, N * 64);
  k_edge_max<8><<<nb((long long)Et * 8, TB), TB, 0, stream>>>(srcI, dstI, aS, aD, mx, Et);
  k_edge_sum<8><<<nb((long long)Et * 8, TB), TB, 0, stream>>>(srcI, dstI, aS, aD, mx, sm, Et);
  k_edge_aggr<8, 8><<<nb((long long)Et * 8, TB), TB, 0, stream>>>(srcI, dstI, aS, aD, mx, sm, bufH, bufO, Et);
  k_bias_relu<64><<<nb((long long)N * 64, TB), TB, 0, stream>>>(bufO, b2, N);

  // ---- Layer 3: 64 -> 1 head x 10 ch ----
  k_gemm_small<<<nb((long long)N * 10, TB), TB, 0, stream>>>(bufO, W3, bufH, N, 64, 10);
  k_attn_logits<1, 10><<<nb(N, TB), TB, 0, stream>>>(bufH, a3s, a3d, aS, aD, mx, sm, N);
  k_zero<<<nb((long long)N * 10, TB), TB, 0, stream>>>(bufO, N * 10);
  k_edge_max<1><<<nb(Et, TB), TB, 0, stream>>>(srcI, dstI, aS, aD, mx, Et);
  k_edge_sum<1><<<nb(Et, TB), TB, 0, stream>>>(srcI, dstI, aS, aD, mx, sm, Et);
  k_edge_aggr<1, 10><<<nb(Et, TB), TB, 0, stream>>>(srcI, dstI, aS, aD, mx, sm, bufH, bufO, Et);
  k_final_logsoftmax<<<nb(N, TB), TB, 0, stream>>>(bufO, b3, y, N);
}